// MultiPosCrossEntropy_48833778155744
// MI455X (gfx1250) — compile-verified
//
#include <hip/hip_runtime.h>
#include <math.h>

// CDNA5 (gfx1250) wave32. Matrix reductions via V_WMMA_F32_16X16X4_F32.
typedef __attribute__((ext_vector_type(2))) float v2f;
typedef __attribute__((ext_vector_type(8))) float v8f;

__device__ __forceinline__ float wave_reduce_max(float v) {
#pragma unroll
  for (int off = 16; off > 0; off >>= 1)
    v = fmaxf(v, __shfl_xor(v, off, 32));
  return v;
}

// Sum 4 f32 values per lane across the whole wave (128 values) using two
// chained V_WMMA_F32_16X16X4_F32 with B = ones:
//   D[i][j] = sum_k A[i][k]  (identical across j), so any column-sum of the
//   16x16 accumulator equals the grand total. The 8 C VGPRs per lane hold
//   rows 0-7 (lanes 0-15) / rows 8-15 (lanes 16-31) of one column; one
//   xor-16 shuffle-add combines the halves. Full fp32, deterministic.
__device__ __forceinline__ float wmma_wave_sum4(float c0, float c1, float c2, float c3) {
  v2f ones; ones[0] = 1.0f; ones[1] = 1.0f;
  v8f acc = {};
  v2f a;
  a[0] = c0; a[1] = c1;
  acc = __builtin_amdgcn_wmma_f32_16x16x4_f32(false, a, false, ones, (short)0, acc, false, false);
  a[0] = c2; a[1] = c3;
  acc = __builtin_amdgcn_wmma_f32_16x16x4_f32(false, a, false, ones, (short)0, acc, false, false);
  float s = acc[0] + acc[1] + acc[2] + acc[3] + acc[4] + acc[5] + acc[6] + acc[7];
  s += __shfl_xor(s, 16, 32);
  return s;  // grand total in every lane
}

// One workgroup (8 waves) per row. C = 1024, 4 elements per thread, held in
// registers across both passes -> each global byte is read exactly once.
__global__ __launch_bounds__(256) void mpce_row_kernel(
    const float* __restrict__ pred, const int* __restrict__ label,
    float* __restrict__ row_loss) {
  constexpr int C = 1024;
  const int row  = blockIdx.x;
  const int tid  = threadIdx.x;
  const int wid  = tid >> 5;
  const int lane = tid & 31;

  // Coalesced 128-bit loads (global_load_b128).
  const float4 p4 = reinterpret_cast<const float4*>(pred  + (size_t)row * C)[tid];
  const int4   l4 = reinterpret_cast<const int4*>(label + (size_t)row * C)[tid];
  const float pv[4] = {p4.x, p4.y, p4.z, p4.w};
  const int   lv[4] = {l4.x, l4.y, l4.z, l4.w};

  // Pass 1: per-side maxima (numerical stabilizers).
  float mn = -INFINITY, mp = -INFINITY;  // max over neg of p; max over pos of -p
#pragma unroll
  for (int k = 0; k < 4; ++k) {
    mn = fmaxf(mn, lv[k] == 0 ? pv[k]  : -INFINITY);
    mp = fmaxf(mp, lv[k] == 1 ? -pv[k] : -INFINITY);
  }
  mn = wave_reduce_max(mn);
  mp = wave_reduce_max(mp);

  __shared__ float s_mn[8], s_mp[8], s_sn[8], s_sp[8];
  if (lane == 0) { s_mn[wid] = mn; s_mp[wid] = mp; }
  __syncthreads();
  float Mn = s_mn[0], Mp = s_mp[0];
#pragma unroll
  for (int i = 1; i < 8; ++i) { Mn = fmaxf(Mn, s_mn[i]); Mp = fmaxf(Mp, s_mp[i]); }

  // Pass 2: masked exp partials (values still in registers).
  float cn[4], cp[4];
#pragma unroll
  for (int k = 0; k < 4; ++k) {
    cn[k] = lv[k] == 0 ? expf(pv[k] - Mn)  : 0.0f;
    cp[k] = lv[k] == 1 ? expf(-pv[k] - Mp) : 0.0f;
  }

  // Wave-level sums via WMMA (EXEC is all-ones here: no divergence at this point).
  const float sn = wmma_wave_sum4(cn[0], cn[1], cn[2], cn[3]);
  const float sp = wmma_wave_sum4(cp[0], cp[1], cp[2], cp[3]);

  if (lane == 0) { s_sn[wid] = sn; s_sp[wid] = sp; }
  __syncthreads();
  if (tid == 0) {
    float Sn = 0.0f, Sp = 0.0f;
#pragma unroll
    for (int i = 0; i < 8; ++i) { Sn += s_sn[i]; Sp += s_sp[i]; }
    // lse = log(sum_neg e^p) + log(sum_pos e^-p); empty side -> -inf -> loss 0.
    const float lse  = Mn + Mp + logf(Sn) + logf(Sp);
    const float loss = fmaxf(lse, 0.0f) + log1pf(expf(-fabsf(lse)));  // logaddexp(lse, 0)
    row_loss[row] = loss;
  }
}

// Single-wave deterministic mean of the 256 per-row losses, again via four
// chained V_WMMA_F32_16X16X4_F32 (each consumes 64 f32 values).
__global__ __launch_bounds__(32) void mpce_mean_kernel(
    const float* __restrict__ row_loss, float* __restrict__ out, float scale) {
  const int lane = threadIdx.x;
  v2f ones; ones[0] = 1.0f; ones[1] = 1.0f;
  v8f acc = {};
#pragma unroll
  for (int c = 0; c < 4; ++c) {
    v2f a;
    a[0] = row_loss[c * 64 + lane];
    a[1] = row_loss[c * 64 + 32 + lane];
    acc = __builtin_amdgcn_wmma_f32_16x16x4_f32(false, a, false, ones, (short)0, acc, false, false);
  }
  float s = acc[0] + acc[1] + acc[2] + acc[3] + acc[4] + acc[5] + acc[6] + acc[7];
  s += __shfl_xor(s, 16, 32);
  if (lane == 0) out[0] = s * scale;
}

extern "C" void kernel_launch(void* const* d_in, const int* in_sizes, int n_in,
                              void* d_out, int out_size, void* d_ws, size_t ws_size,
                              hipStream_t stream) {
  const float* pred  = (const float*)d_in[0];
  const int*   label = (const int*)d_in[1];
  float*       out   = (float*)d_out;
  float*       rloss = (float*)d_ws;  // 256 floats of scratch

  constexpr int C = 1024;
  const int N = in_sizes[0] / C;  // 256 for this problem

  mpce_row_kernel<<<N, 256, 0, stream>>>(pred, label, rloss);
  // LOSS_WEIGHT = 1.0; mean reduction = sum / N
  mpce_mean_kernel<<<1, 32, 0, stream>>>(rloss, out, 1.0f / (float)N);
}